// ShoeboxToRIR_22265110462477
// MI455X (gfx1250) — compile-verified
//
#include <hip/hip_runtime.h>
#include <stdint.h>

#define USE_ASYNC_WB 1

static constexpr int   kRirLen  = 24000;
static constexpr int   kA       = 42;          // images per axis: 2*(2*10+1)
static constexpr int   kNImg    = kA * kA * kA;
static constexpr int   kTaps    = 81;
static constexpr int   kChunk   = 27;          // 81 = 3 * 27 tap chunks
static constexpr int   kHalf    = 40;
static constexpr int   kMaxOrd  = 10;
static constexpr int   kMaxKept = 2048;        // exact kept count is 1561
static constexpr float kPi      = 3.14159265358979323846f;
static constexpr float kSrOverC = 139.9416909620991f;     // 48000 / 343
static constexpr float kInv4Pi  = 0.07957747154594767f;   // 1/(4*pi)
static constexpr float kInvPi   = 0.3183098861837907f;    // 1/pi
static constexpr float kCd      = 0.9970658282f;          // cos(pi/41)
static constexpr float kSd      = 0.0765489045f;          // sin(pi/41)

__global__ __launch_bounds__(1024)
void ShoeboxToRIR_22265110462477_kernel(const float* __restrict__ in,
                                        float* __restrict__ out)
{
    // Dynamic LDS: full per-batch RIR accumulator (96 KB < 320 KB WGP LDS).
    // Declared float4 so the base is 16B-aligned for b128 async stores.
    extern __shared__ float4 dyn_smem[];
    float* rir = reinterpret_cast<float*>(dyn_smem);

    __shared__ float axd[3][kA];       // per-axis (image - mic) coordinate
    __shared__ int   axord[kA];        // per-axis wall-hit order
    __shared__ float betap[kMaxOrd + 1];
    __shared__ int   klist[kMaxKept];  // compacted kept images
    __shared__ int   nkept;

    const int b   = blockIdx.x;
    const int tid = threadIdx.x;
    const int nth = blockDim.x;

    if (tid == 0) nkept = 0;

    // ---- zero the LDS RIR accumulator ----
    for (int j = tid; j < kRirLen; j += nth) rir[j] = 0.0f;

    // ---- per-batch geometry (uniform -> scalar loads) ----
    float room[3], mic[3], src[3];
#pragma unroll
    for (int k = 0; k < 3; ++k) {
        float r = in[b * 9 + k] * 10.0f;
        room[k] = r;
        mic[k]  = in[b * 9 + 3 + k] * r;
        src[k]  = in[b * 9 + 6 + k] * r;
    }

    // ---- per-axis image tables: coord = sign*src + 2n*room - mic ----
    if (tid < 3 * kA) {
        int axis = tid / kA;
        int a    = tid - axis * kA;
        int p    = (a >= 21) ? 1 : 0;            // ravel order: p-major
        int n    = (a - 21 * p) - 10;            // n in [-10, 10]
        float sgn = p ? -1.0f : 1.0f;
        axd[axis][a] = sgn * src[axis] + 2.0f * (float)n * room[axis] - mic[axis];
        if (axis == 0) {
            int d0 = n - p; if (d0 < 0) d0 = -d0;
            int d1 = n;     if (d1 < 0) d1 = -d1;
            axord[a] = d0 + d1;                  // |n-p| + |n|
        }
    }
    if (tid <= kMaxOrd) {                        // beta^k table
        float v = 1.0f;
        for (int q = 0; q < tid; ++q) v *= 0.9f;
        betap[tid] = v;
    }
    __syncthreads();

    // ---- phase A: scan 42^3 lattice, compact survivors into LDS worklist ----
    for (int i = tid; i < kNImg; i += nth) {
        int az = i % kA;
        int r1 = i / kA;
        int ay = r1 % kA;
        int ax = r1 / kA;
        int tot = axord[ax] + axord[ay] + axord[az];
        if (tot <= kMaxOrd) {
            int pos = atomicAdd(&nkept, 1);      // ds_add_rtn_u32
            if (pos < kMaxKept)
                klist[pos] = az | (ay << 6) | (ax << 12) | (tot << 18);
        }
    }
    __syncthreads();

    // ---- phase B: dense tap evaluation, full EXEC per wave ----
    // Work unit = (kept image, 27-tap chunk): ~4700 units over 1024 threads.
    const int nunits = nkept * 3;
    for (int u = tid; u < nunits; u += nth) {
        int img   = u / 3;                       // magic-number div
        int chunk = u - img * 3;
        int code  = klist[img];
        int az  = code & 63;
        int ay  = (code >> 6) & 63;
        int ax  = (code >> 12) & 63;
        int tot = code >> 18;

        float dx = axd[0][ax], dy = axd[1][ay], dz = axd[2][az];
        float dist = sqrtf(dx * dx + dy * dy + dz * dz);
        float amp  = __fdividef(betap[tot] * kInv4Pi, dist);

        float tau  = dist * kSrOverC;            // fractional delay (samples)
        float i0f  = floorf(tau);
        float frac = tau - i0f;
        int   base = (int)i0f + kHalf;           // idx = i0 + HALF + t

        // sin(pi*(t-40-frac)) = -(-1)^t * sin(pi*frac): one transcendental/unit
        float sp = __sinf(kPi * frac);
        int   t0 = chunk * kChunk;
        float salt = ((t0 & 1) ? sp : -sp) * kInvPi;   // alternating numerator/pi

        // Hann window cos(pi*tt/41): seed at chunk start, rotate by pi/41 per tap
        float cw, sw;
        __sincosf(((float)(t0 - kHalf) - frac) * (kPi / 41.0f), &sw, &cw);

        for (int t = t0; t < t0 + kChunk; ++t) {
            int   idx = base + t;
            float tt  = (float)(t - kHalf) - frac;
            if ((unsigned)idx < (unsigned)kRirLen && fabsf(tt) <= 40.0f) {
                float sinc = (tt == 0.0f) ? 1.0f : __fdividef(salt, tt);
                float win  = 0.5f + 0.5f * cw;
                atomicAdd(&rir[idx], amp * sinc * win);   // ds_add_f32
            }
            float cn = cw * kCd - sw * kSd;      // rotate window angle
            sw = sw * kCd + cw * kSd;
            cw = cn;
            salt = -salt;
        }
    }

    __syncthreads();

    // ---- origin output ----
    if (tid == 0) {
        float ddx = mic[0] - src[0], ddy = mic[1] - src[1], ddz = mic[2] - src[2];
        out[(size_t)gridDim.x * kRirLen + b] =
            40.0f + sqrtf(ddx * ddx + ddy * ddy + ddz * ddz) * kSrOverC;
    }

    // ---- writeback: async LDS -> HBM (ASYNCcnt path, no VGPR staging) ----
    float* gout = out + (size_t)b * kRirLen;
#if USE_ASYNC_WB
    {
        unsigned long long gbase = (unsigned long long)(uintptr_t)gout;
        for (int j = tid; j < kRirLen / 4; j += nth) {
            unsigned goff  = (unsigned)(j * 16);
            // low 32 bits of a generic shared pointer == LDS byte address
            unsigned laddr = (unsigned)(uintptr_t)(rir + j * 4);
            asm volatile("global_store_async_from_lds_b128 %0, %1, %2"
                         :: "v"(goff), "v"(laddr), "s"(gbase)
                         : "memory");
        }
        asm volatile("s_wait_asynccnt 0" ::: "memory");
    }
#else
    {
        const float4* s4 = reinterpret_cast<const float4*>(rir);
        float4*       g4 = reinterpret_cast<float4*>(gout);
        for (int j = tid; j < kRirLen / 4; j += nth) g4[j] = s4[j];
    }
#endif
}

extern "C" void kernel_launch(void* const* d_in, const int* in_sizes, int n_in,
                              void* d_out, int out_size, void* d_ws, size_t ws_size,
                              hipStream_t stream) {
    const float* in  = (const float*)d_in[0];
    float*       out = (float*)d_out;
    const int B = in_sizes[0] / 9;               // [B, 9] input
    ShoeboxToRIR_22265110462477_kernel<<<dim3(B), dim3(1024),
                                         kRirLen * sizeof(float), stream>>>(in, out);
}